// MaskingActivationLayer_2147483648371
// MI455X (gfx1250) — compile-verified
//
#include <hip/hip_runtime.h>
#include <hip/hip_bf16.h>

// Problem constants (from reference): B=256, S=8192, L=S-1=8191, TOK=11, 8 classes.
#define B_ROWS 256
#define S_LEN  8192
#define L_LEN  8191
#define TOK    11

// ---------------------------------------------------------------------------
// Pass 2 (hot pass, placed first so the disasm snippet shows it):
// fully parallel masked softmax, one thread per (b, l).
//   mask bits (bit j = class j):
//     t=0:0x02  t=1:0x06  t=2:0x10  t=3:0xF8  t=7:0x80  t>=8:0x00
//     t in {4,5,6}: !seen5 -> 0x20, else !seen6 -> 0x40, else 0xF8
//   where seen5 = (l >= first5[b]), seen6 = (l >= first6[b]).
// logits = scores + (mask?0:-1e9); probs = softmax(logits).
// Traffic/thread: 32B scores in + 32B probs out + 4B strided type word.
// ---------------------------------------------------------------------------
__global__ void __launch_bounds__(256)
masked_softmax_kernel(const int* __restrict__ song,
                      const float4* __restrict__ scores,
                      const int* __restrict__ firsts,
                      float4* __restrict__ out) {
    const int b = blockIdx.y;
    const int l = blockIdx.x * 256 + (int)threadIdx.x;
    if (l >= L_LEN) return;

    const long long ti = ((long long)b * S_LEN + l) * TOK;
    // gfx1250 prefetch path: kick the strided song cacheline as early as
    // possible (only 4 of every 44 bytes are useful; latency-bound stream).
    __builtin_prefetch(&song[ti], 0, 3);           // -> global_prefetch_b8

    const long long si = ((long long)b * L_LEN + l) * 2;  // float4 units
    const float4 s0 = scores[si + 0];              // global_load_b128
    const float4 s1 = scores[si + 1];

    const int f5 = firsts[2 * b + 0];              // uniform per block
    const int f6 = firsts[2 * b + 1];
    const int t  = song[ti];

    // Branchless mask: compute both candidates, v_cndmask-select.
    const unsigned base = ((unsigned)t < 8u)
            ? (unsigned)((0x80000000F8100602ULL >> (t * 8)) & 0xFFu)
            : 0u;                                   // bytes LE: 02 06 10 F8 00 00 00 80
    const unsigned note = (l < f5) ? 0x20u : ((l < f6) ? 0x40u : 0xF8u);
    const unsigned m    = ((unsigned)(t - 4) <= 2u) ? note : base;

    float v[8] = { s0.x, s0.y, s0.z, s0.w, s1.x, s1.y, s1.z, s1.w };
    float mx = -3.402823466e38f;
#pragma unroll
    for (int j = 0; j < 8; ++j) {
        v[j] += ((m >> j) & 1u) ? 0.0f : -1.0e9f;
        mx = fmaxf(mx, v[j]);
    }
    float sum = 0.0f;
#pragma unroll
    for (int j = 0; j < 8; ++j) {
        v[j] = __expf(v[j] - mx);                  // v_exp_f32
        sum += v[j];
    }
    const float inv = 1.0f / sum;

    float4 o0, o1;
    o0.x = v[0] * inv; o0.y = v[1] * inv; o0.z = v[2] * inv; o0.w = v[3] * inv;
    o1.x = v[4] * inv; o1.y = v[5] * inv; o1.z = v[6] * inv; o1.w = v[7] * inv;
    out[si + 0] = o0;                              // global_store_b128
    out[si + 1] = o1;
}

// ---------------------------------------------------------------------------
// Pass 1: per batch row, find first index of a type-5 and type-6 token.
// Inclusive prefix-OR collapses to two scalars per row. One block per row,
// LDS atomicMin + uniform early-exit (types uniform 0..8 -> expected ~1 seg).
// ---------------------------------------------------------------------------
__global__ void __launch_bounds__(256)
find_first_kernel(const int* __restrict__ song, int* __restrict__ firsts) {
    __shared__ int f5, f6;
    const int b = blockIdx.x;
    if (threadIdx.x == 0) { f5 = 0x7FFFFFFF; f6 = 0x7FFFFFFF; }
    __syncthreads();

    const long long base = (long long)b * S_LEN * TOK;
    for (int seg = 0; seg < L_LEN; seg += 256) {
        const int pos = seg + (int)threadIdx.x;
        if (pos < L_LEN) {
            const int t = song[base + (long long)pos * TOK];
            if (t == 5) atomicMin(&f5, pos);
            if (t == 6) atomicMin(&f6, pos);
        }
        __syncthreads();
        // Values uniform across block after the barrier -> uniform break.
        if (f5 != 0x7FFFFFFF && f6 != 0x7FFFFFFF) break;
        __syncthreads();  // protect LDS reads above from next iter's atomics
    }
    if (threadIdx.x == 0) {
        firsts[2 * b + 0] = f5;
        firsts[2 * b + 1] = f6;
    }
}

extern "C" void kernel_launch(void* const* d_in, const int* in_sizes, int n_in,
                              void* d_out, int out_size, void* d_ws, size_t ws_size,
                              hipStream_t stream) {
    const int*    song   = (const int*)d_in[0];    // [B, S, TOK] int32
    const float4* scores = (const float4*)d_in[1]; // [B, L, 8] float32
    float4*       out    = (float4*)d_out;         // [B, L, 8] float32
    int*          firsts = (int*)d_ws;             // [B, 2] scratch (512 ints)

    (void)in_sizes; (void)n_in; (void)out_size; (void)ws_size;

    // Pass 1: first occurrence of type-5 / type-6 per row (writes all of the
    // workspace region it uses; no init needed despite 0xAA poison).
    find_first_kernel<<<dim3(B_ROWS), dim3(256), 0, stream>>>(song, firsts);

    // Pass 2: streaming masked softmax, one thread per (b, l).
    dim3 grid((L_LEN + 255) / 256, B_ROWS);
    masked_softmax_kernel<<<grid, dim3(256), 0, stream>>>(song, scores, firsts, out);
}